// DGLSAGE_42691974922961
// MI455X (gfx1250) — compile-verified
//
#include <hip/hip_runtime.h>

typedef __attribute__((ext_vector_type(2))) float v2f;
typedef __attribute__((ext_vector_type(8))) float v8f;

// ---------------------------------------------------------------------------
// Zero a float buffer (grid-stride).
// ---------------------------------------------------------------------------
__global__ void zero_f32(float* __restrict__ p, long n) {
  long i = (long)blockIdx.x * blockDim.x + threadIdx.x;
  long stride = (long)gridDim.x * blockDim.x;
  for (; i < n; i += stride) p[i] = 0.0f;
}

// ---------------------------------------------------------------------------
// In-degree: deg[dst[e]] += 1.0  (L2-resident f32 atomics)
// ---------------------------------------------------------------------------
__global__ void degree_kernel(const int* __restrict__ dst, float* __restrict__ deg, int E) {
  int e = blockIdx.x * blockDim.x + threadIdx.x;
  if (e < E) atomicAdd(&deg[dst[e]], 1.0f);
}

// ---------------------------------------------------------------------------
// Edge-parallel scatter-add: one wave32 per edge, 128 feats, lane moves 4
// consecutive floats (float4 gather from L2-resident x, 4x atomic f32 adds).
// ---------------------------------------------------------------------------
__global__ void scatter_add_kernel(const float* __restrict__ x, const int* __restrict__ src,
                                   const int* __restrict__ dst, float* __restrict__ agg, int E) {
  int wid = (int)((blockIdx.x * (unsigned)blockDim.x + threadIdx.x) >> 5);
  int lane = threadIdx.x & 31;
  if (wid >= E) return;
  int s = src[wid];
  int d = dst[wid];
  const float4 v = *(const float4*)(x + (size_t)s * 128 + lane * 4);
  float* a = agg + (size_t)d * 128 + lane * 4;
  atomicAdd(a + 0, v.x);
  atomicAdd(a + 1, v.y);
  atomicAdd(a + 2, v.z);
  atomicAdd(a + 3, v.w);
}

// ---------------------------------------------------------------------------
// hn[i] = agg[i] / max(deg[node], 1)   (feat width fixed at 128)
// ---------------------------------------------------------------------------
__global__ void normalize_kernel(const float* __restrict__ agg, const float* __restrict__ deg,
                                 float* __restrict__ hn, long total) {
  long i = (long)blockIdx.x * blockDim.x + threadIdx.x;
  if (i >= total) return;
  long node = i >> 7;  // / 128
  hn[i] = agg[i] / fmaxf(deg[node], 1.0f);
}

// ---------------------------------------------------------------------------
// Fused SAGE linear:  Out = maybe_relu( Xs @ Wself + Xn @ Wneigh + bias )
// Xs, Xn: [Nnodes x K] row-major (K=128). W: [K x F] row-major. F in {64,128}.
// One 16x16 output tile per wave via V_WMMA_F32_16X16X4_F32 (f32 in/accum).
//
// A layout (16x4 f32): lanes 0-15 hold M=lane, {K=k0+0, k0+1}; lanes 16-31
// hold M=lane-16, {K=k0+2, k0+3}  -> one contiguous float2 load per lane.
// B layout (4x16 f32): VGPR0 = rows {k0+0 | k0+2}, VGPR1 = rows {k0+1 | k0+3},
// N=lane%16 striped across lanes.
// C/D layout (8 VGPRs): VGPR r -> (M=r, N=lane) lanes 0-15, (M=8+r) lanes 16-31.
// ---------------------------------------------------------------------------
__global__ void sage_linear_wmma(const float* __restrict__ Xs, const float* __restrict__ Xn,
                                 const float* __restrict__ Wself, const float* __restrict__ Wneigh,
                                 const float* __restrict__ bias, float* __restrict__ Out,
                                 int Nnodes, int K, int F, int do_relu) {
  const int wid = threadIdx.x >> 5;
  const int lane = threadIdx.x & 31;
  const int tiles_n = F >> 4;                 // # of 16-wide output tiles
  const int m_tiles_per_block = 8 / tiles_n;  // 8 waves per 256-thread block
  const int m_tile = blockIdx.x * m_tiles_per_block + wid / tiles_n;
  const int n0 = (wid % tiles_n) << 4;
  const int m0 = m_tile << 4;
  if (m0 >= Nnodes) return;                   // wave-uniform: EXEC stays all-1s

  const int lane_lo = lane & 15;
  const int khalf = (lane >> 4) << 1;         // 0 for lanes 0-15, 2 for 16-31
  const size_t rowA = (size_t)(m0 + lane_lo) * K;

  v8f c = {};
  for (int k0 = 0; k0 < K; k0 += 4) {
    const int k = k0 + khalf;
    // self path
    v2f a = *(const v2f*)(Xs + rowA + k);
    v2f b;
    b.x = Wself[(size_t)k * F + n0 + lane_lo];
    b.y = Wself[(size_t)(k + 1) * F + n0 + lane_lo];
    c = __builtin_amdgcn_wmma_f32_16x16x4_f32(false, a, false, b, (short)0, c, false, false);
    // neighbor path (accumulate into same C)
    v2f a2 = *(const v2f*)(Xn + rowA + k);
    v2f b2;
    b2.x = Wneigh[(size_t)k * F + n0 + lane_lo];
    b2.y = Wneigh[(size_t)(k + 1) * F + n0 + lane_lo];
    c = __builtin_amdgcn_wmma_f32_16x16x4_f32(false, a2, false, b2, (short)0, c, false, false);
  }

  const float bv = bias[n0 + lane_lo];
  const int mbase = m0 + ((lane >> 4) << 3);  // lanes 16-31 write rows M=8..15
#pragma unroll
  for (int r = 0; r < 8; ++r) {
    float v = c[r] + bv;
    if (do_relu) v = fmaxf(v, 0.0f);
    Out[(size_t)(mbase + r) * F + n0 + lane_lo] = v;
  }
}

// ---------------------------------------------------------------------------
// Host-side orchestration (all launches on `stream`, graph-capture safe).
// ---------------------------------------------------------------------------
extern "C" void kernel_launch(void* const* d_in, const int* in_sizes, int n_in,
                              void* d_out, int out_size, void* d_ws, size_t ws_size,
                              hipStream_t stream) {
  (void)in_sizes; (void)n_in; (void)out_size; (void)ws_size;

  const float* x   = (const float*)d_in[0];
  const int*   src = (const int*)d_in[1];
  const int*   dst = (const int*)d_in[2];
  const float* W1s = (const float*)d_in[3];
  const float* W1n = (const float*)d_in[4];
  const float* b1  = (const float*)d_in[5];
  const float* W2s = (const float*)d_in[6];
  const float* W2n = (const float*)d_in[7];
  const float* b2  = (const float*)d_in[8];
  float* out = (float*)d_out;

  const int N = 50000, E = 800000, H = 128, F2 = 64;
  const long NH = (long)N * H;

  // Workspace layout (floats): deg[N] | agg[N*H] | hn[N*H] | h[N*H]  (~77 MB)
  float* ws  = (float*)d_ws;
  float* deg = ws;
  float* agg = ws + N;
  float* hn  = agg + NH;
  float* h   = hn + NH;

  const int ZB = 2048, T = 256;
  const int edge_waves_blocks = (E * 32 + T - 1) / T;  // one wave per edge
  const int norm_blocks = (int)((NH + T - 1) / T);

  // ---- Layer 1 ----
  zero_f32<<<ZB, T, 0, stream>>>(deg, (long)N + NH);  // deg + agg are adjacent
  degree_kernel<<<(E + T - 1) / T, T, 0, stream>>>(dst, deg, E);
  scatter_add_kernel<<<edge_waves_blocks, T, 0, stream>>>(x, src, dst, agg, E);
  normalize_kernel<<<norm_blocks, T, 0, stream>>>(agg, deg, hn, NH);
  // F=128 -> tiles_n=8, 1 m-tile per block -> grid = N/16 = 3125
  sage_linear_wmma<<<N / 16, T, 0, stream>>>(x, hn, W1s, W1n, b1, h, N, H, H, 1);

  // ---- Layer 2 (deg reused; agg re-zeroed) ----
  zero_f32<<<ZB, T, 0, stream>>>(agg, NH);
  scatter_add_kernel<<<edge_waves_blocks, T, 0, stream>>>(h, src, dst, agg, E);
  normalize_kernel<<<norm_blocks, T, 0, stream>>>(agg, deg, hn, NH);
  // F=64 -> tiles_n=4, 2 m-tiles per block -> grid = ceil(3125/2) = 1563
  sage_linear_wmma<<<(N / 16 + 1) / 2, T, 0, stream>>>(h, hn, W2s, W2n, b2, out, N, H, F2, 0);
}